// DoubleConv1_37941741093013
// MI455X (gfx1250) — compile-verified
//
#include <hip/hip_runtime.h>

typedef __attribute__((ext_vector_type(2))) float v2f;
typedef __attribute__((ext_vector_type(8))) float v8f;

#define B_    4
#define CIN   32
#define COUT  64
#define H_    192
#define W_    192
#define NOFF  9          // 3x3 kernel positions
#define KDEF  (CIN*NOFF) // 288
#define KC2   (COUT*9)   // 576
#define EPSV  1e-5f

// ---------------------------------------------------------------------------
// Kernel 1: offset conv  off = conv2d(x, w_off, pad=1) + b_off   (B,18,H,W)
// ---------------------------------------------------------------------------
__global__ __launch_bounds__(192)
void offset_conv_kernel(const float* __restrict__ x,
                        const float* __restrict__ w_off,
                        const float* __restrict__ b_off,
                        float* __restrict__ off)
{
    const int w  = threadIdx.x;               // 0..191
    const int h  = blockIdx.x % H_;
    const int oc = (blockIdx.x / H_) % 18;
    const int b  = blockIdx.x / (H_ * 18);

    float acc = b_off[oc];
    for (int ic = 0; ic < CIN; ++ic) {
        const float* xb = x + ((size_t)(b * CIN + ic) * H_) * W_;
        const float* wk = w_off + (oc * CIN + ic) * 9;
#pragma unroll
        for (int ky = 0; ky < 3; ++ky) {
            int hy = h - 1 + ky;
            if ((unsigned)hy >= (unsigned)H_) continue;
#pragma unroll
            for (int kx = 0; kx < 3; ++kx) {
                int wx = w - 1 + kx;
                if ((unsigned)wx >= (unsigned)W_) continue;
                acc = fmaf(xb[hy * W_ + wx], wk[ky * 3 + kx], acc);
            }
        }
    }
    off[(size_t)blockIdx.x * W_ + w] = acc;   // blockIdx = (b*18+oc)*H_+h
}

// ---------------------------------------------------------------------------
// Kernel 2: deformable sampling + GEMM (WMMA f32 16x16x4) + BN1 + ReLU -> y1
// One workgroup: 32 consecutive pixels of one image row, all 64 out channels.
// ---------------------------------------------------------------------------
__global__ __launch_bounds__(256)
void deform_wmma_kernel(const float* __restrict__ x,
                        const float* __restrict__ off,
                        const float* __restrict__ w_def,   // (64, 288) k=c*9+n
                        const float* __restrict__ gamma1,
                        const float* __restrict__ beta1,
                        const float* __restrict__ mean1,
                        const float* __restrict__ var1,
                        float* __restrict__ y1)
{
    __shared__ float xoff[32][KDEF];          // 36 KB
    __shared__ int   s_iy0[32 * NOFF], s_ix0[32 * NOFF];
    __shared__ int   s_iy1[32 * NOFF], s_ix1[32 * NOFF];
    __shared__ float s_wlt[32 * NOFF], s_wrb[32 * NOFF];
    __shared__ float s_wlb[32 * NOFF], s_wrt[32 * NOFF];

    const int tid = threadIdx.x;
    const int P   = blockIdx.x * 32;
    const int b   = P / (H_ * W_);
    const int rem = P % (H_ * W_);
    const int h   = rem / W_;
    const int w0  = rem % W_;
    const float HpM1 = (float)(H_ + 1);       // Hp-1
    const float WpM1 = (float)(W_ + 1);       // Wp-1

    // Phase A: bilinear metadata for 32 pixels x 9 kernel positions
    for (int idx = tid; idx < 32 * NOFF; idx += 256) {
        int p = idx / NOFF, n = idx % NOFF;
        float offy = off[((size_t)(b * 18 + n) * H_ + h) * W_ + (w0 + p)];
        float offx = off[((size_t)(b * 18 + 9 + n) * H_ + h) * W_ + (w0 + p)];
        float py = offy + (float)(h + 1) + (float)(n / 3 - 1);
        float px = offx + (float)(w0 + p + 1) + (float)(n % 3 - 1);
        float fy = floorf(py), fx = floorf(px);
        float qy0 = fminf(fmaxf(fy,        0.f), HpM1);
        float qx0 = fminf(fmaxf(fx,        0.f), WpM1);
        float qy1 = fminf(fmaxf(fy + 1.f,  0.f), HpM1);
        float qx1 = fminf(fmaxf(fx + 1.f,  0.f), WpM1);
        float pyc = fminf(fmaxf(py,        0.f), HpM1);
        float pxc = fminf(fmaxf(px,        0.f), WpM1);
        float wy0 = 1.f + (qy0 - pyc);
        float wy1 = 1.f - (qy1 - pyc);
        float wx0 = 1.f + (qx0 - pxc);
        float wx1 = 1.f - (qx1 - pxc);
        s_iy0[idx] = (int)qy0; s_ix0[idx] = (int)qx0;
        s_iy1[idx] = (int)qy1; s_ix1[idx] = (int)qx1;
        s_wlt[idx] = wy0 * wx0; s_wrb[idx] = wy1 * wx1;
        s_wlb[idx] = wy0 * wx1; s_wrt[idx] = wy1 * wx0;
    }
    __syncthreads();

    // Phase B: gather/blend -> xoff[pixel][c*9+n]
    for (int idx = tid; idx < 32 * KDEF; idx += 256) {
        int p = idx / KDEF;
        int k = idx % KDEF;
        int c = k / NOFF, n = k % NOFF;
        int mi = p * NOFF + n;
        const float* xc = x + ((size_t)(b * CIN + c) * H_) * W_;
        int iy0 = s_iy0[mi], ix0 = s_ix0[mi];
        int iy1 = s_iy1[mi], ix1 = s_ix1[mi];
        auto samp = [&](int iy, int ix) -> float {
            // (iy,ix) are padded-map coords; interior iff 1..H / 1..W
            if (iy >= 1 && iy <= H_ && ix >= 1 && ix <= W_)
                return xc[(iy - 1) * W_ + (ix - 1)];
            return 0.f;
        };
        float v = s_wlt[mi] * samp(iy0, ix0) + s_wrb[mi] * samp(iy1, ix1)
                + s_wlb[mi] * samp(iy0, ix1) + s_wrt[mi] * samp(iy1, ix0);
        xoff[p][k] = v;
    }
    __syncthreads();

    // Phase C: 8 waves -> 2 pixel-tiles x 4 channel-tiles of 16x16
    const int lane  = tid & 31;
    const int wv    = tid >> 5;
    const int ptile = wv >> 2;
    const int obase = (wv & 3) * 16;
    const int mlo   = lane & 15;
    const int khi   = (lane >> 4) << 1;       // 0 or 2

    v8f acc = {0.f, 0.f, 0.f, 0.f, 0.f, 0.f, 0.f, 0.f};
    const float* wrow = w_def + (size_t)(obase + mlo) * KDEF;
    for (int k = 0; k < KDEF; k += 4) {
        int kk = k + khi;
        v2f a, bb;
        a.x  = xoff[ptile * 16 + mlo][kk];
        a.y  = xoff[ptile * 16 + mlo][kk + 1];
        bb.x = wrow[kk];
        bb.y = wrow[kk + 1];
        acc = __builtin_amdgcn_wmma_f32_16x16x4_f32(
                  false, a, false, bb, (short)0, acc, false, false);
    }

    // Epilogue: BN1 + ReLU, store y1 (B,64,H,W)
    const int o  = obase + mlo;
    float sc = gamma1[o] * rsqrtf(var1[o] + EPSV);
    float bi = beta1[o] - mean1[o] * sc;
    const int mbase = ptile * 16 + ((lane >> 4) << 3);
    float* yrow = y1 + ((size_t)(b * COUT + o) * H_ + h) * W_ + w0;
#pragma unroll
    for (int r = 0; r < 8; ++r) {
        float v = fmaxf(acc[r] * sc + bi, 0.f);
        yrow[mbase + r] = v;
    }
}

// ---------------------------------------------------------------------------
// Kernel 3: conv2 3x3 (64->64) via implicit-im2col WMMA + bias + BN2 + ReLU
// ---------------------------------------------------------------------------
__global__ __launch_bounds__(256)
void conv2_wmma_kernel(const float* __restrict__ y1,
                       const float* __restrict__ w2,     // (64, 576) k=c*9+ky*3+kx
                       const float* __restrict__ b2,
                       const float* __restrict__ gamma2,
                       const float* __restrict__ beta2,
                       const float* __restrict__ mean2,
                       const float* __restrict__ var2,
                       float* __restrict__ out)
{
    __shared__ float tile[COUT * 3 * 34];     // 26 KB: [c][row 0..2][col 0..33]

    const int tid = threadIdx.x;
    const int P   = blockIdx.x * 32;
    const int b   = P / (H_ * W_);
    const int rem = P % (H_ * W_);
    const int h   = rem / W_;
    const int w0  = rem % W_;

    // Phase A: stage y1 halo tile (zero-padded)
    for (int idx = tid; idx < COUT * 3 * 34; idx += 256) {
        int c  = idx / 102;
        int rr = idx % 102;
        int r  = rr / 34, col = rr % 34;
        int hy = h - 1 + r, wx = w0 - 1 + col;
        float v = 0.f;
        if ((unsigned)hy < (unsigned)H_ && (unsigned)wx < (unsigned)W_)
            v = y1[((size_t)(b * COUT + c) * H_ + hy) * W_ + wx];
        tile[idx] = v;
    }
    __syncthreads();

    // Phase B: WMMA over K = 576
    const int lane  = tid & 31;
    const int wv    = tid >> 5;
    const int ptile = wv >> 2;
    const int obase = (wv & 3) * 16;
    const int mlo   = lane & 15;
    const int khi   = (lane >> 4) << 1;

    v8f acc = {0.f, 0.f, 0.f, 0.f, 0.f, 0.f, 0.f, 0.f};
    const float* wrow = w2 + (size_t)(obase + mlo) * KC2;
    const int pcol = ptile * 16 + mlo;
    for (int k = 0; k < KC2; k += 4) {
        int k0 = k + khi;
        int k1 = k0 + 1;
        int c0 = k0 / 9, rm0 = k0 % 9;
        int c1 = k1 / 9, rm1 = k1 % 9;
        v2f a, bb;
        a.x  = tile[c0 * 102 + (rm0 / 3) * 34 + pcol + (rm0 % 3)];
        a.y  = tile[c1 * 102 + (rm1 / 3) * 34 + pcol + (rm1 % 3)];
        bb.x = wrow[k0];
        bb.y = wrow[k1];
        acc = __builtin_amdgcn_wmma_f32_16x16x4_f32(
                  false, a, false, bb, (short)0, acc, false, false);
    }

    // Epilogue: + b2, BN2, ReLU
    const int o  = obase + mlo;
    float sc = gamma2[o] * rsqrtf(var2[o] + EPSV);
    float bi = (b2[o] - mean2[o]) * sc + beta2[o];
    const int mbase = ptile * 16 + ((lane >> 4) << 3);
    float* orow = out + ((size_t)(b * COUT + o) * H_ + h) * W_ + w0;
#pragma unroll
    for (int r = 0; r < 8; ++r) {
        float v = fmaxf(acc[r] * sc + bi, 0.f);
        orow[mbase + r] = v;
    }
}

// ---------------------------------------------------------------------------
extern "C" void kernel_launch(void* const* d_in, const int* in_sizes, int n_in,
                              void* d_out, int out_size, void* d_ws, size_t ws_size,
                              hipStream_t stream)
{
    const float* x      = (const float*)d_in[0];
    const float* w_off  = (const float*)d_in[1];
    const float* b_off  = (const float*)d_in[2];
    const float* w_def  = (const float*)d_in[3];
    const float* gamma1 = (const float*)d_in[4];
    const float* beta1  = (const float*)d_in[5];
    const float* mean1  = (const float*)d_in[6];
    const float* var1   = (const float*)d_in[7];
    const float* w2     = (const float*)d_in[8];
    const float* b2     = (const float*)d_in[9];
    const float* gamma2 = (const float*)d_in[10];
    const float* beta2  = (const float*)d_in[11];
    const float* mean2  = (const float*)d_in[12];
    const float* var2   = (const float*)d_in[13];

    float* off = (float*)d_ws;                              // (4,18,192,192)
    float* y1  = off + (size_t)B_ * 18 * H_ * W_;           // (4,64,192,192)

    offset_conv_kernel<<<B_ * 18 * H_, 192, 0, stream>>>(x, w_off, b_off, off);

    const int nblk = (B_ * H_ * W_) / 32;                   // 4608 full blocks
    deform_wmma_kernel<<<nblk, 256, 0, stream>>>(
        x, off, w_def, gamma1, beta1, mean1, var1, y1);

    conv2_wmma_kernel<<<nblk, 256, 0, stream>>>(
        y1, w2, b2, gamma2, beta2, mean2, var2, (float*)d_out);
}